// AltDiff_47244640256493
// MI455X (gfx1250) — compile-verified
//
#include <hip/hip_runtime.h>
#include <hip/hip_bf16.h>

typedef __attribute__((ext_vector_type(2))) float v2f;
typedef __attribute__((ext_vector_type(8))) float v8f;

#define BATCH 8192
#define NF 512
#define NH 128
#define NC 10
#define NEQ 32
#define NINEQ 64
#define QPEN 0.1f
#define NITER 50

// workspace layout (float offsets); B matrices stored PAIR-INTERLEAVED:
//   Bp[(k>>1)*2N + 2n + (k&1)] = B[k][n]
#define WS_BK   0                        // Kinv^T as B operand [128][128] interleaved
#define WS_BGT  (WS_BK + NH*NH)          // G^T  [128][64] interleaved
#define WS_BAT  (WS_BGT + NH*NINEQ)      // A^T  [128][32] interleaved
#define WS_AB   (WS_BAT + NH*NEQ)        // A·b  [128]
#define WS_R0   (WS_AB + NH)             // r0 = Ab - relu(fc1)  [8192][128]

// padded LDS leading dims (even; dword strides coprime-ish with 64 banks)
#define LDZ 132
#define LDSN 68
#define LDL 36

__device__ __forceinline__ v8f wmma4(v2f a, v2f b, v8f c) {
    // D = A(16x4) x B(4x16) + C, fp32 in / fp32 accumulate
    return __builtin_amdgcn_wmma_f32_16x16x4_f32(false, a, false, b,
                                                 (short)0, c, false, false);
}

__device__ __forceinline__ v2f ld2(const float* p) {
    return *(const v2f*)p;   // 8B-aligned pair -> ds_load_b64 / global b64
}

// ---------------------------------------------------------------------------
// prep: K = 0.1 I + A'A + G'G ; Kinv via Gauss-Jordan (SPD) ; pack B operands
// ---------------------------------------------------------------------------
__global__ __launch_bounds__(128)
void prep_kernel(const float* __restrict__ Ag, const float* __restrict__ Gg,
                 const float* __restrict__ bg, float* __restrict__ ws) {
    extern __shared__ float sm[];
    float* Ksm = sm;               // 128 x 129 (padded)
    float* Ism = sm + 128 * 129;   // 128 x 129
    const int tid = threadIdx.x;

    for (int idx = tid; idx < NH * NH; idx += 128) {
        int i = idx >> 7, j = idx & 127;
        float acc = (i == j) ? QPEN : 0.0f;
        for (int e = 0; e < NEQ; ++e)   acc += Ag[e * NH + i] * Ag[e * NH + j];
        for (int e = 0; e < NINEQ; ++e) acc += Gg[e * NH + i] * Gg[e * NH + j];
        Ksm[i * 129 + j] = acc;
        Ism[i * 129 + j] = (i == j) ? 1.0f : 0.0f;
    }
    __syncthreads();

    const int r = tid;  // one row per thread
    for (int c = 0; c < NH; ++c) {
        if (r == c) {
            float pinv = 1.0f / Ksm[c * 129 + c];
            for (int j = 0; j < NH; ++j) {
                Ksm[c * 129 + j] *= pinv;
                Ism[c * 129 + j] *= pinv;
            }
        }
        __syncthreads();
        if (r != c) {
            float f = Ksm[r * 129 + c];
            for (int j = 0; j < NH; ++j) {
                Ksm[r * 129 + j] -= f * Ksm[c * 129 + j];
                Ism[r * 129 + j] -= f * Ism[c * 129 + j];
            }
        }
        __syncthreads();
    }

    // BK pair-interleaved: element (k,n) = Kinv[n][k]
    for (int idx = tid; idx < NH * NH; idx += 128) {
        int k = idx >> 7, n = idx & 127;
        ws[WS_BK + (k >> 1) * (2 * NH) + 2 * n + (k & 1)] = Ism[n * 129 + k];
    }
    // BGT pair-interleaved: element (k,e) = G[e][k]   (K=128, N=64)
    for (int idx = tid; idx < NH * NINEQ; idx += 128) {
        int k = idx / NINEQ, e = idx % NINEQ;
        ws[WS_BGT + (k >> 1) * (2 * NINEQ) + 2 * e + (k & 1)] = Gg[e * NH + k];
    }
    // BAT pair-interleaved: element (k,e) = A[e][k]   (K=128, N=32)
    for (int idx = tid; idx < NH * NEQ; idx += 128) {
        int k = idx / NEQ, e = idx % NEQ;
        ws[WS_BAT + (k >> 1) * (2 * NEQ) + 2 * e + (k & 1)] = Ag[e * NH + k];
    }
    for (int k = tid; k < NH; k += 128) {
        float acc = 0.0f;
        for (int e = 0; e < NEQ; ++e) acc += bg[e] * Ag[e * NH + k];
        ws[WS_AB + k] = acc;
    }
}

// ---------------------------------------------------------------------------
// fc1: r0 = Ab - relu(x @ W1^T + b1)  via WMMA, 64 rows/block, 4 k-chunks
// (one-time ~1 GFLOP; left in the simple scalar-fragment form)
// ---------------------------------------------------------------------------
__global__ __launch_bounds__(128)
void fc1_kernel(const float* __restrict__ x, const float* __restrict__ w1,
                const float* __restrict__ b1, const float* __restrict__ ws_ab,
                float* __restrict__ r0) {
    extern __shared__ float sm[];
    float* xs = sm;             // 64 x 128
    float* Bc = sm + 64 * 128;  // 128 x 129 (padded)
    const int tid = threadIdx.x;
    const int lane = tid & 31, wave = tid >> 5;
    const int lo = lane & 15, hi = lane >> 4;
    const int row0 = blockIdx.x * 64;
    const int rb = wave * 16;

    v8f acc[8];
    for (int nt = 0; nt < 8; ++nt)
        for (int q = 0; q < 8; ++q) acc[nt][q] = 0.0f;

    for (int kc = 0; kc < 4; ++kc) {
        for (int idx = tid; idx < 64 * 128; idx += 128) {
            int rr = idx >> 7, j = idx & 127;
            xs[rr * 128 + j] = x[(size_t)(row0 + rr) * NF + kc * 128 + j];
        }
        for (int idx = tid; idx < 128 * 128; idx += 128) {
            int n = idx >> 7, k = idx & 127;
            Bc[k * 129 + n] = w1[(size_t)n * NF + kc * 128 + k];
        }
        __syncthreads();
        for (int nt = 0; nt < 8; ++nt) {
            v8f c = acc[nt];
            for (int kt = 0; kt < 32; ++kt) {
                int k = kt * 4 + hi * 2;
                v2f a = ld2(&xs[(rb + lo) * 128 + k]);
                v2f b;
                b.x = Bc[k * 129 + nt * 16 + lo];
                b.y = Bc[(k + 1) * 129 + nt * 16 + lo];
                c = wmma4(a, b, c);
            }
            acc[nt] = c;
        }
        __syncthreads();
    }
    for (int nt = 0; nt < 8; ++nt) {
        int n = nt * 16 + lo;
        float bias = b1[n];
        float abv = ws_ab[n];
        for (int r = 0; r < 8; ++r) {
            int m = row0 + rb + r + hi * 8;
            float p = acc[nt][r] + bias;
            p = p > 0.0f ? p : 0.0f;
            r0[(size_t)m * NH + n] = abv - p;
        }
    }
}

// ---------------------------------------------------------------------------
// ADMM: 50 iterations fully LDS-resident; each wave owns 16 rows (no barriers
// inside the loop). State L = -lambda so all GEMMs accumulate with '+'.
// All B operands pair-interleaved -> one ds_load_b64 per WMMA operand.
// ---------------------------------------------------------------------------
__global__ __launch_bounds__(128)
void admm_kernel(const float* __restrict__ Gg, const float* __restrict__ Ag,
                 const float* __restrict__ hg, const float* __restrict__ bg,
                 const float* __restrict__ ws, const float* __restrict__ w2,
                 const float* __restrict__ b2, float* __restrict__ out) {
    extern __shared__ float sm[];
    float* BK  = sm;                   // 128x128 interleaved
    float* BG  = BK  + NH * NH;        // 64x128  interleaved
    float* BGT = BG  + NINEQ * NH;     // 128x64  interleaved
    float* BA  = BGT + NH * NINEQ;     // 32x128  interleaved
    float* BAT = BA  + NEQ * NH;       // 128x32  interleaved
    float* r0t = BAT + NH * NEQ;       // 64x128
    float* zr  = r0t + 64 * NH;        // 64xLDZ   rhs
    float* zz  = zr  + 64 * LDZ;       // 64xLDZ   z
    float* ss  = zz  + 64 * LDZ;       // 64xLDSN  s
    float* nn  = ss  + 64 * LDSN;      // 64xLDSN  nu
    float* LL  = nn  + 64 * LDSN;      // 64xLDL   -lambda
    float* hs  = LL  + 64 * LDL;       // 64
    float* bs  = hs  + NINEQ;          // 32

    const int tid = threadIdx.x;
    const int lane = tid & 31, wave = tid >> 5;
    const int lo = lane & 15, hi = lane >> 4;
    const int rb = wave * 16;
    const int row0 = blockIdx.x * 64;

    // stage shared matrices (BK/BGT/BAT already interleaved in ws)
    for (int i = tid; i < NH * NH; i += 128)    BK[i]  = ws[WS_BK + i];
    for (int idx = tid; idx < NINEQ * NH; idx += 128) {       // BG = G, interleave
        int k = idx >> 7, n = idx & 127;
        BG[(k >> 1) * (2 * NH) + 2 * n + (k & 1)] = Gg[idx];
    }
    for (int i = tid; i < NH * NINEQ; i += 128) BGT[i] = ws[WS_BGT + i];
    for (int idx = tid; idx < NEQ * NH; idx += 128) {         // BA = A, interleave
        int k = idx >> 7, n = idx & 127;
        BA[(k >> 1) * (2 * NH) + 2 * n + (k & 1)] = Ag[idx];
    }
    for (int i = tid; i < NH * NEQ; i += 128)   BAT[i] = ws[WS_BAT + i];
    for (int i = tid; i < 64 * NH; i += 128)    r0t[i] = ws[WS_R0 + (size_t)row0 * NH + i];
    for (int i = tid; i < 64 * LDSN; i += 128) { ss[i] = 0.0f; nn[i] = 0.0f; }
    for (int i = tid; i < 64 * LDL; i += 128)   LL[i] = 0.0f;
    for (int i = tid; i < NINEQ; i += 128)      hs[i] = hg[i];
    for (int i = tid; i < NEQ; i += 128)        bs[i] = bg[i];
    __syncthreads();

    const int nA = 2 * (0) + 2 * lo;  // (unused placeholder to keep indices clear)
    (void)nA;

    for (int it = 0; it < NITER; ++it) {
        // rhs = r0 + L@A + ((h-s)-nu)@G
        for (int nt = 0; nt < 8; ++nt) {
            const int n2 = 2 * (nt * 16 + lo);
            v8f c;
            for (int r = 0; r < 8; ++r)
                c[r] = r0t[(rb + r + hi * 8) * NH + nt * 16 + lo];
            for (int kt = 0; kt < 8; ++kt) {             // K = NEQ = 32
                int k = kt * 4 + hi * 2;
                v2f a = ld2(&LL[(rb + lo) * LDL + k]);
                v2f b = ld2(&BA[(k >> 1) * (2 * NH) + n2]);
                c = wmma4(a, b, c);
            }
            for (int kt = 0; kt < 16; ++kt) {            // K = NINEQ = 64
                int k = kt * 4 + hi * 2;
                v2f hv = ld2(&hs[k]);
                v2f sv = ld2(&ss[(rb + lo) * LDSN + k]);
                v2f nv = ld2(&nn[(rb + lo) * LDSN + k]);
                v2f a = hv - sv - nv;
                v2f b = ld2(&BG[(k >> 1) * (2 * NH) + n2]);
                c = wmma4(a, b, c);
            }
            for (int r = 0; r < 8; ++r)
                zr[(rb + r + hi * 8) * LDZ + nt * 16 + lo] = c[r];
        }
        // z = rhs @ Kinv^T
        for (int nt = 0; nt < 8; ++nt) {
            const int n2 = 2 * (nt * 16 + lo);
            v8f c;
            for (int r = 0; r < 8; ++r) c[r] = 0.0f;
            for (int kt = 0; kt < 32; ++kt) {            // K = NH = 128
                int k = kt * 4 + hi * 2;
                v2f a = ld2(&zr[(rb + lo) * LDZ + k]);
                v2f b = ld2(&BK[(k >> 1) * (2 * NH) + n2]);
                c = wmma4(a, b, c);
            }
            for (int r = 0; r < 8; ++r)
                zz[(rb + r + hi * 8) * LDZ + nt * 16 + lo] = c[r];
        }
        // w = z @ G^T ; s = relu(h-w-nu) ; nu = relu(w+nu-h)
        for (int nt = 0; nt < 4; ++nt) {
            const int n2 = 2 * (nt * 16 + lo);
            v8f c;
            for (int r = 0; r < 8; ++r) c[r] = 0.0f;
            for (int kt = 0; kt < 32; ++kt) {
                int k = kt * 4 + hi * 2;
                v2f a = ld2(&zz[(rb + lo) * LDZ + k]);
                v2f b = ld2(&BGT[(k >> 1) * (2 * NINEQ) + n2]);
                c = wmma4(a, b, c);
            }
            for (int r = 0; r < 8; ++r) {
                int m = rb + r + hi * 8, j = nt * 16 + lo;
                float t = hs[j] - c[r] - nn[m * LDSN + j];
                ss[m * LDSN + j] = fmaxf(t, 0.0f);
                nn[m * LDSN + j] = fmaxf(-t, 0.0f);
            }
        }
        // v = z @ A^T ; L += b - v   (L = -lambda)
        for (int nt = 0; nt < 2; ++nt) {
            const int n2 = 2 * (nt * 16 + lo);
            v8f c;
            for (int r = 0; r < 8; ++r) c[r] = 0.0f;
            for (int kt = 0; kt < 32; ++kt) {
                int k = kt * 4 + hi * 2;
                v2f a = ld2(&zz[(rb + lo) * LDZ + k]);
                v2f b = ld2(&BAT[(k >> 1) * (2 * NEQ) + n2]);
                c = wmma4(a, b, c);
            }
            for (int r = 0; r < 8; ++r) {
                int m = rb + r + hi * 8, e = nt * 16 + lo;
                LL[m * LDL + e] += bs[e] - c[r];
            }
        }
    }
    __syncthreads();

    // fc2 + log_softmax, one thread per row
    if (tid < 64) {
        float logits[NC];
        for (int cc = 0; cc < NC; ++cc) {
            float acc = b2[cc];
            for (int j = 0; j < NH; ++j) acc += zz[tid * LDZ + j] * w2[cc * NH + j];
            logits[cc] = acc;
        }
        float mx = logits[0];
        for (int cc = 1; cc < NC; ++cc) mx = fmaxf(mx, logits[cc]);
        float se = 0.0f;
        for (int cc = 0; cc < NC; ++cc) se += __expf(logits[cc] - mx);
        float lse = mx + __logf(se);
        for (int cc = 0; cc < NC; ++cc)
            out[(size_t)(row0 + tid) * NC + cc] = logits[cc] - lse;
    }
}

extern "C" void kernel_launch(void* const* d_in, const int* in_sizes, int n_in,
                              void* d_out, int out_size, void* d_ws, size_t ws_size,
                              hipStream_t stream) {
    const float* x  = (const float*)d_in[0];
    const float* w1 = (const float*)d_in[1];
    const float* b1 = (const float*)d_in[2];
    const float* w2 = (const float*)d_in[3];
    const float* b2 = (const float*)d_in[4];
    const float* G  = (const float*)d_in[5];
    const float* h  = (const float*)d_in[6];
    const float* A  = (const float*)d_in[7];
    const float* b  = (const float*)d_in[8];
    float* ws  = (float*)d_ws;
    float* out = (float*)d_out;
    (void)in_sizes; (void)n_in; (void)out_size; (void)ws_size;

    const size_t smem_prep = 2u * 128u * 129u * sizeof(float);              // 132 KB
    const size_t smem_fc1  = (64u * 128u + 128u * 129u) * sizeof(float);    // ~97 KB
    const size_t smem_admm =
        (size_t)(NH * NH + NINEQ * NH + NH * NINEQ + NEQ * NH + NH * NEQ +
                 64 * NH + 64 * LDZ * 2 + 64 * LDSN * 2 + 64 * LDL +
                 NINEQ + NEQ) * sizeof(float);                              // ~301 KB

    prep_kernel<<<1, 128, smem_prep, stream>>>(A, G, b, ws);
    fc1_kernel<<<BATCH / 64, 128, smem_fc1, stream>>>(x, w1, b1, ws + WS_AB, ws + WS_R0);
    admm_kernel<<<BATCH / 64, 128, smem_admm, stream>>>(G, A, h, b, ws, w2, b2, out);
}